// LSTM_17755394802270
// MI455X (gfx1250) — compile-verified
//
#include <hip/hip_runtime.h>

#define B_TOTAL 16384
#define T_STEPS 512
#define F_IN    3
#define H_DIM   4

typedef __attribute__((ext_vector_type(2))) float v2f;
typedef __attribute__((ext_vector_type(8))) float v8f;

__device__ __forceinline__ float fast_tanh(float x) {
#if __has_builtin(__builtin_amdgcn_tanhf)
    return __builtin_amdgcn_tanhf(x);
#elif __has_builtin(__builtin_amdgcn_tanh_f32)
    return __builtin_amdgcn_tanh_f32(x);
#else
    // tanh(x) = (1 - e^{-2x}) / (1 + e^{-2x}), e^{-2x} = exp2(-2x*log2(e))
    float e = __builtin_amdgcn_exp2f(-2.885390082f * x);
    return (1.0f - e) * __builtin_amdgcn_rcpf(1.0f + e);
#endif
}

__device__ __forceinline__ float fast_sigmoid(float x) {
    return fmaf(0.5f, fast_tanh(0.5f * x), 0.5f);
}

// ---------------------------------------------------------------------------
// Kernel 1: the sequential LSTM scan. One thread = one batch element.
// Weights are wave-uniform -> scalar loads; h/c state lives in VGPRs.
// x is streamed as 16B-aligned float4 (3 per 4 timesteps), prefetched.
// ---------------------------------------------------------------------------
__global__ void __launch_bounds__(256) lstm_scan_kernel(
    const float* __restrict__ x,
    const float* __restrict__ Wx,   // [3,16]
    const float* __restrict__ Wh,   // [4,16]
    const float* __restrict__ bias, // [16]
    float* __restrict__ hout)       // [B,4]
{
    const int b = blockIdx.x * 256 + threadIdx.x;

    float wx[3][16], wh[4][16], bb[16];
#pragma unroll
    for (int j = 0; j < 16; ++j) {
        bb[j] = bias[j];
#pragma unroll
        for (int k = 0; k < 3; ++k) wx[k][j] = Wx[k * 16 + j];
#pragma unroll
        for (int k = 0; k < 4; ++k) wh[k][j] = Wh[k * 16 + j];
    }

    float h[4] = {0.f, 0.f, 0.f, 0.f};
    float c[4] = {0.f, 0.f, 0.f, 0.f};

    // Each thread owns a contiguous T*F = 1536-float (6144 B) slice.
    const float4* xp = reinterpret_cast<const float4*>(x) +
                       (size_t)b * (T_STEPS * F_IN / 4);
    float4 p0 = xp[0], p1 = xp[1], p2 = xp[2];

    for (int t4 = 0; t4 < T_STEPS / 4; ++t4) {
        float xs[12] = {p0.x, p0.y, p0.z, p0.w,
                        p1.x, p1.y, p1.z, p1.w,
                        p2.x, p2.y, p2.z, p2.w};
        if (t4 + 1 < T_STEPS / 4) {  // software prefetch next 4 steps
            p0 = xp[(t4 + 1) * 3 + 0];
            p1 = xp[(t4 + 1) * 3 + 1];
            p2 = xp[(t4 + 1) * 3 + 2];
        }
#pragma unroll
        for (int s = 0; s < 4; ++s) {
            const float x0 = xs[3 * s + 0];
            const float x1 = xs[3 * s + 1];
            const float x2 = xs[3 * s + 2];
            float z[16];
#pragma unroll
            for (int j = 0; j < 16; ++j) {
                z[j] = fmaf(x0, wx[0][j],
                       fmaf(x1, wx[1][j],
                       fmaf(x2, wx[2][j],
                       fmaf(h[0], wh[0][j],
                       fmaf(h[1], wh[1][j],
                       fmaf(h[2], wh[2][j],
                       fmaf(h[3], wh[3][j], bb[j])))))));
            }
#pragma unroll
            for (int k = 0; k < 4; ++k) {
                float ig = fast_sigmoid(z[k]);
                float fg = fast_sigmoid(z[4 + k]);
                float gg = fast_tanh(z[8 + k]);
                float og = fast_sigmoid(z[12 + k]);
                c[k] = fmaf(fg, c[k], ig * gg);
                h[k] = og * fast_tanh(c[k]);
            }
        }
    }
    reinterpret_cast<float4*>(hout)[b] = make_float4(h[0], h[1], h[2], h[3]);
}

// ---------------------------------------------------------------------------
// Kernel 2: head. hid = relu(h_final @ W1 + b1); out = softmax(hid @ W2 + b2).
// h_final @ W1 is exactly one V_WMMA_F32_16X16X4_F32 per 16 batch rows:
//   A (16x4 f32): lane m=lane&15, VGPR0 = K(lane<16?0:2), VGPR1 = K(+1)
//   B (4x16 f32): lane n=lane&15, VGPR0 = K(lane<16?0:2) row of W1
//   C: b1 broadcast down the rows. ReLU on C-layout regs, stage via LDS.
// ---------------------------------------------------------------------------
__global__ void __launch_bounds__(256) head_kernel(
    const float* __restrict__ hfin, // [B,4]
    const float* __restrict__ W1,   // [4,16]
    const float* __restrict__ b1,   // [16]
    const float* __restrict__ W2,   // [16,5]
    const float* __restrict__ b2,   // [5]
    float* __restrict__ out)        // [B,5]
{
    __shared__ float hid_lds[8][16][16]; // one 16x16 tile per wave (8 KB)

    const int  lane = threadIdx.x & 31;
    const int  wave = threadIdx.x >> 5;
    const int  n    = lane & 15;
    const bool hi   = lane >= 16;
    const int  row0 = blockIdx.x * 128 + wave * 16;

    // A operand: 16 rows of h_final, K=4
    float4 hv = reinterpret_cast<const float4*>(hfin)[row0 + n];
    v2f a;
    a.x = hi ? hv.z : hv.x;  // K = 0 or 2
    a.y = hi ? hv.w : hv.y;  // K = 1 or 3

    // B operand: W1 rows striped K across lane halves
    v2f bm;
    bm.x = hi ? W1[2 * 16 + n] : W1[0 * 16 + n];
    bm.y = hi ? W1[3 * 16 + n] : W1[1 * 16 + n];

    // C: bias broadcast (same value for every row m, column n = lane&15)
    float bv = b1[n];
    v8f cc;
#pragma unroll
    for (int v = 0; v < 8; ++v) cc[v] = bv;

    v8f d = __builtin_amdgcn_wmma_f32_16x16x4_f32(
        /*neg_a=*/false, a, /*neg_b=*/false, bm,
        /*c_mod=*/(short)0, cc, /*reuse_a=*/false, /*reuse_b=*/false);

    // ReLU + un-stripe C layout (VGPR v -> row m, lane -> col n) via LDS
#pragma unroll
    for (int v = 0; v < 8; ++v) {
        int m = hi ? (v + 8) : v;
        hid_lds[wave][m][n] = fmaxf(d[v], 0.0f);
    }
    __syncthreads();

    // 128 rows per block; threads 0..127 each finish one row (16x5 + softmax)
    if (threadIdx.x < 128) {
        const int w = threadIdx.x >> 4;
        const int r = threadIdx.x & 15;
        float hid[16];
#pragma unroll
        for (int j = 0; j < 16; ++j) hid[j] = hid_lds[w][r][j];

        float logit[5];
#pragma unroll
        for (int q = 0; q < 5; ++q) {
            float acc = b2[q];
#pragma unroll
            for (int j = 0; j < 16; ++j) acc = fmaf(hid[j], W2[j * 5 + q], acc);
            logit[q] = acc;
        }
        float mx = logit[0];
#pragma unroll
        for (int q = 1; q < 5; ++q) mx = fmaxf(mx, logit[q]);
        float p[5], ssum = 0.f;
#pragma unroll
        for (int q = 0; q < 5; ++q) {
            p[q] = __builtin_amdgcn_exp2f((logit[q] - mx) * 1.44269504f);
            ssum += p[q];
        }
        float inv = __builtin_amdgcn_rcpf(ssum);
        const int row = blockIdx.x * 128 + w * 16 + r;
#pragma unroll
        for (int q = 0; q < 5; ++q) out[row * 5 + q] = p[q] * inv;
    }
}

extern "C" void kernel_launch(void* const* d_in, const int* in_sizes, int n_in,
                              void* d_out, int out_size, void* d_ws, size_t ws_size,
                              hipStream_t stream)
{
    const float* x  = (const float*)d_in[0];
    const float* Wx = (const float*)d_in[1];
    const float* Wh = (const float*)d_in[2];
    const float* b  = (const float*)d_in[3];
    const float* W1 = (const float*)d_in[4];
    const float* b1 = (const float*)d_in[5];
    const float* W2 = (const float*)d_in[6];
    const float* b2 = (const float*)d_in[7];
    float* out  = (float*)d_out;
    float* hfin = (float*)d_ws; // B*4 floats = 256 KB scratch

    lstm_scan_kernel<<<B_TOTAL / 256, 256, 0, stream>>>(x, Wx, Wh, b, hfin);
    head_kernel<<<B_TOTAL / 128, 256, 0, stream>>>(hfin, W1, b1, W2, b2, out);
}